// Decoder_50345606643971
// MI455X (gfx1250) — compile-verified
//
#include <hip/hip_runtime.h>
#include <hip/hip_bf16.h>
#include <cstddef>

typedef float v2f __attribute__((ext_vector_type(2)));
typedef float v8f __attribute__((ext_vector_type(8)));

constexpr int B   = 64;
constexpr int L   = 2048;
constexpr int H   = 512;
constexpr int E   = 512;
constexpr int V   = 50257;
constexpr int D   = 2 * H;     // 1024
constexpr int IN3 = 3 * H;     // 1536 == E + D

// ---------------------------------------------------------------- attn fill
__global__ void attn_fill_kernel(float* __restrict__ attn) {
    int i = blockIdx.x * 256 + threadIdx.x;          // B*L = 131072 total
    attn[i] = 1.0f / (float)L;
}

// --------------------------------------------------- encoder_out mean: pass1
__global__ __launch_bounds__(256)
void enc_partial_kernel(const float* __restrict__ enc, float* __restrict__ partial) {
    const int d  = blockIdx.x * 256 + threadIdx.x;   // 0..1023
    const int lz = blockIdx.y;                       // 0..7
    const int b  = blockIdx.z;                       // 0..63
    const float* p = enc + ((size_t)b * L + (size_t)lz * (L / 8)) * D + d;
    float s = 0.0f;
    #pragma unroll 8
    for (int l = 0; l < L / 8; ++l) s += p[(size_t)l * D];
    partial[((size_t)b * 8 + lz) * D + d] = s;
}

// --------------------------------------------------- encoder_out mean: pass2
__global__ void ctx_finish_kernel(const float* __restrict__ partial,
                                  float* __restrict__ xt,       // [B, IN3]
                                  float* __restrict__ concat2)  // [B, IN3]
{
    const int d = blockIdx.x * 256 + threadIdx.x;    // 0..1023
    const int b = blockIdx.y;
    float s = 0.0f;
    #pragma unroll
    for (int lz = 0; lz < 8; ++lz) s += partial[((size_t)b * 8 + lz) * D + d];
    const float c = s * (1.0f / (float)L);
    xt[(size_t)b * IN3 + E + d]      = c;   // xt = [xe, context]
    concat2[(size_t)b * IN3 + H + d] = c;   // concat2 = [h_new, context]
}

// ---------------------------------------------------------------- embedding
__global__ void embed_kernel(const int* __restrict__ x, const float* __restrict__ emb,
                             float* __restrict__ xt) {
    const int e = blockIdx.x * 256 + threadIdx.x;    // 0..511
    const int b = blockIdx.y;
    xt[(size_t)b * IN3 + e] = emb[(size_t)x[b] * E + e];
}

// ------------------------------------------------ GEMV: out[b,j] = vec[b]·W[j]
template <int K>
__global__ __launch_bounds__(256)
void gemv_kernel(const float* __restrict__ vec,   // [B, K]
                 const float* __restrict__ W,     // [IN3, K] row-major
                 const float* __restrict__ bias,  // [IN3]
                 float* __restrict__ out)         // [B, IN3]
{
    const int b    = blockIdx.y;
    const int j    = blockIdx.x * 8 + (threadIdx.x >> 5);  // wave per output row
    const int lane = threadIdx.x & 31;
    const float* wr = W   + (size_t)j * K;
    const float* v  = vec + (size_t)b * K;
    float s = 0.0f;
    #pragma unroll 4
    for (int k = lane; k < K; k += 32) s += wr[k] * v[k];
    #pragma unroll
    for (int off = 16; off > 0; off >>= 1) s += __shfl_xor(s, off, 32);
    if (lane == 0) out[(size_t)b * IN3 + j] = s + bias[j];
}

// --------------------------------------------------------------- GRU gates
__global__ void gru_gate_kernel(const float* __restrict__ gi, const float* __restrict__ gh,
                                const float* __restrict__ h0,
                                float* __restrict__ out_hidden,  // [B, H]
                                float* __restrict__ concat2)     // [B, IN3]
{
    const int b = blockIdx.y;
    const int j = blockIdx.x * 256 + threadIdx.x;    // 0..511
    const size_t g = (size_t)b * IN3 + j;
    const float ir = gi[g], iz = gi[g + H], inx = gi[g + 2 * H];
    const float hr = gh[g], hz = gh[g + H], hnv = gh[g + 2 * H];
    const float r = 1.0f / (1.0f + expf(-(ir + hr)));
    const float z = 1.0f / (1.0f + expf(-(iz + hz)));
    const float n = tanhf(inx + r * hnv);
    const float h0v = h0[(size_t)b * H + j];
    const float hn = (1.0f - z) * n + z * h0v;
    out_hidden[(size_t)b * H + j] = hn;
    concat2[(size_t)b * IN3 + j]  = hn;
}

// ---------------------------------------- logits GEMM via V_WMMA_F32_16X16X4
// C[64, V] = A[64, IN3] * Bw[IN3, V] + bias.  Block: 8 waves, M-tile 16 rows,
// each wave owns a 16x16 N-tile (128 cols/block; blocks step 256 cols). A chunk staged in LDS.
constexpr int KC      = 512;   // K chunk staged in LDS
constexpr int ASTRIDE = 516;   // padded LDS row stride (floats) -> no bank conflicts

__global__ __launch_bounds__(256)
void logits_wmma_kernel(const float* __restrict__ A,    // concat2 [64, IN3]
                        const float* __restrict__ Bw,   // out_w   [IN3, V]
                        const float* __restrict__ bias, // out_b   [V]
                        float* __restrict__ Cout)       // logits  [64, V]
{
    __shared__ float aLds[16 * ASTRIDE];
    const int tid   = threadIdx.x;
    const int wave  = tid >> 5;
    const int lane  = tid & 31;
    const int mbase = blockIdx.x * 16;                 // 0,16,32,48
    const int n0    = blockIdx.y * 256 + wave * 16;
    const int m     = lane & 15;
    const int half  = lane >> 4;                       // 0 | 1
    const int kk0   = half << 1;                       // A/B frag: K = kk0, kk0+1
    const int col   = n0 + (lane & 15);
    const int colc  = (col < V) ? col : (V - 1);       // clamp tail loads in-bounds
    const bool cok  = (col < V);

    v8f acc = {};
    for (int kc0 = 0; kc0 < IN3; kc0 += KC) {
        // stage A[mbase:+16, kc0:+KC) into LDS (float4, conflict-light)
        for (int i = tid; i < 16 * (KC / 4); i += 256) {
            const int row = i / (KC / 4);
            const int kq  = i % (KC / 4);
            const float4 v4 = *(const float4*)(A + (size_t)(mbase + row) * IN3 + kc0 + kq * 4);
            *(float4*)(&aLds[row * ASTRIDE + kq * 4]) = v4;
        }
        __syncthreads();

        const float* pB = Bw + (size_t)(kc0 + kk0) * V + colc;
        const float* pA = &aLds[m * ASTRIDE + kk0];
        #pragma unroll 4
        for (int k = 0; k < KC; k += 4) {
            const float2 av = *(const float2*)(pA + k);   // ds_load_b64
            v2f a; a.x = av.x; a.y = av.y;
            v2f bfrag; bfrag.x = pB[0]; bfrag.y = pB[(size_t)V];
            pB += (size_t)4 * V;
            acc = __builtin_amdgcn_wmma_f32_16x16x4_f32(
                    false, a, false, bfrag, (short)0, acc, false, false);
        }
        __syncthreads();
    }

    if (cok) {
        const float bv = bias[col];
        const int r0 = mbase + half * 8;   // D layout: VGPR i -> rows i and i+8
        #pragma unroll
        for (int i = 0; i < 8; ++i)
            Cout[(size_t)(r0 + i) * V + col] = acc[i] + bv;
    }
}

// ------------------------------------------------ per-row log-sum-exp (det.)
__global__ __launch_bounds__(256)
void rowstats_kernel(const float* __restrict__ logits, float* __restrict__ lse) {
    __shared__ float sm[256], ss[256];
    const int b = blockIdx.x, t = threadIdx.x;
    const float* row = logits + (size_t)b * V;
    float mx = -3.402823466e38f, sum = 0.0f;
    for (int v = t; v < V; v += 256) {
        const float xv = row[v];
        if (xv > mx) { sum *= expf(mx - xv); mx = xv; }
        sum += expf(xv - mx);
    }
    sm[t] = mx; ss[t] = sum;
    __syncthreads();
    for (int off = 128; off > 0; off >>= 1) {
        if (t < off) {
            const float m1 = sm[t], s1 = ss[t];
            const float m2 = sm[t + off], s2 = ss[t + off];
            const float mm = (m1 > m2) ? m1 : m2;
            sm[t] = mm;
            ss[t] = s1 * expf(m1 - mm) + s2 * expf(m2 - mm);
        }
        __syncthreads();
    }
    if (t == 0) lse[b] = sm[0] + logf(ss[0]);
}

__global__ void logp_kernel(const float* __restrict__ logits, const float* __restrict__ lse,
                            float* __restrict__ out_logp) {
    const int col = blockIdx.x * 256 + threadIdx.x;
    const int b   = blockIdx.y;
    if (col < V) out_logp[(size_t)b * V + col] = logits[(size_t)b * V + col] - lse[b];
}

// ---------------------------------------------------------------------------
extern "C" void kernel_launch(void* const* d_in, const int* in_sizes, int n_in,
                              void* d_out, int out_size, void* d_ws, size_t ws_size,
                              hipStream_t stream) {
    (void)in_sizes; (void)n_in; (void)out_size; (void)ws_size;

    const int*   x     = (const int*)  d_in[0];
    const float* h0    = (const float*)d_in[1];   // [1,B,H] -> [B,H]
    const float* enc   = (const float*)d_in[2];   // [B,L,D]
    /* use_cuda d_in[3], attn_w d_in[5], attn_b d_in[6] are dead code */
    const float* emb   = (const float*)d_in[4];   // [V,E]
    const float* w_ih  = (const float*)d_in[7];   // [3H, E+D]
    const float* w_hh  = (const float*)d_in[8];   // [3H, H]
    const float* b_ih  = (const float*)d_in[9];
    const float* b_hh  = (const float*)d_in[10];
    const float* out_w = (const float*)d_in[11];  // [3H, V]
    const float* out_b = (const float*)d_in[12];

    float* out_logp   = (float*)d_out;                       // [B,V]
    float* out_hidden = out_logp + (size_t)B * V;            // [1,B,H]
    float* out_attn   = out_hidden + (size_t)B * H;          // [B,1,L]

    float* ws      = (float*)d_ws;
    float* partial = ws;                                     // 8*B*D
    float* xt      = partial + (size_t)8 * B * D;            // B*IN3
    float* gi      = xt + (size_t)B * IN3;                   // B*IN3
    float* gh      = gi + (size_t)B * IN3;                   // B*IN3
    float* concat2 = gh + (size_t)B * IN3;                   // B*IN3
    float* logits  = concat2 + (size_t)B * IN3;              // B*V
    float* lse     = logits + (size_t)B * V;                 // B

    // 1) constant attention weights (softmax over singleton -> uniform 1/L)
    attn_fill_kernel<<<(B * L) / 256, 256, 0, stream>>>(out_attn);
    // 2) embedding -> xt[:, :E]
    embed_kernel<<<dim3(E / 256, B), 256, 0, stream>>>(x, emb, xt);
    // 3) context = mean_l encoder_out  (two-phase, deterministic)
    enc_partial_kernel<<<dim3(D / 256, 8, B), 256, 0, stream>>>(enc, partial);
    ctx_finish_kernel<<<dim3(D / 256, B), 256, 0, stream>>>(partial, xt, concat2);
    // 4) GRU gate pre-activations
    gemv_kernel<IN3><<<dim3(IN3 / 8, B), 256, 0, stream>>>(xt, w_ih, b_ih, gi);
    gemv_kernel<H>  <<<dim3(IN3 / 8, B), 256, 0, stream>>>(h0, w_hh, b_hh, gh);
    // 5) gates -> h_new (hidden output + concat2[:, :H])
    gru_gate_kernel<<<dim3(H / 256, B), 256, 0, stream>>>(gi, gh, h0, out_hidden, concat2);
    // 6) logits = concat2 @ out_w + out_b   (WMMA f32; grid.x fastest so the
    //    4 M-tiles of one N-chunk run together -> out_w streamed ~once via L2)
    logits_wmma_kernel<<<dim3(B / 16, (V + 255) / 256), 256, 0, stream>>>(
        concat2, out_w, out_b, logits);
    // 7) log-softmax
    rowstats_kernel<<<B, 256, 0, stream>>>(logits, lse);
    logp_kernel<<<dim3((V + 255) / 256, B), 256, 0, stream>>>(logits, lse, out_logp);
}